// DNC_73246372266024
// MI455X (gfx1250) — compile-verified
//
#include <hip/hip_runtime.h>
#include <hip/hip_bf16.h>

// ---------------- problem constants ----------------
#define N_TOT   262144
#define WC      128
#define RHEADS  4
#define OUTD    512
#define IND     512
#define IFACE_D 919          // full iface width (row stride of W_interface)
#define IFACE_N 645          // entries we actually need: 4*128 + 4 + 128 + 1
#define C_DIM   1431
#define Z_DIM   5724         // 4*C

typedef unsigned long long u64;
typedef __attribute__((ext_vector_type(2))) float v2f;
typedef __attribute__((ext_vector_type(8))) float v8f;

__device__ __forceinline__ float sigf(float x) { return 1.f / (1.f + __expf(-x)); }

__device__ __forceinline__ void atomicMaxF(float* addr, float val) {
    unsigned* ua = (unsigned*)addr;
    unsigned old = *ua;
    while (true) {
        float of = __uint_as_float(old);
        if (of >= val) break;
        unsigned assumed = old;
        old = atomicCAS(ua, assumed, __float_as_uint(val));
        if (old == assumed) break;
    }
}

// ---------------- init: h/c copy, colmax/colsum reset, xw = x@Wd + bd -------
__global__ void k_init(const float* __restrict__ x, const float* __restrict__ Wd,
                       const float* __restrict__ bd, const float* __restrict__ h0,
                       const float* __restrict__ c0, float* h, float* c, float* xw,
                       float* colmax, float* colsum) {
    int g = blockIdx.x * 256 + threadIdx.x;
    if (g < C_DIM) { h[g] = h0[g]; c[g] = c0[g]; }
    if (g < 5) { colmax[g] = -3.4e38f; colsum[g] = 0.f; }
    if (blockIdx.x == 5 && threadIdx.x < WC) {   // tiny GEMV: 128 cols x 512 dot
        int j = threadIdx.x;
        float acc = bd[j];
        for (int i = 0; i < IND; ++i) acc = fmaf(x[i], Wd[i * WC + j], acc);
        xw[j] = acc;
    }
}

// ---------------- LSTM z = xt@K + h@R + b  (split-K=4 over 64 cols/block) ---
__global__ __launch_bounds__(256) void k_zmat(const float* __restrict__ xt,
                                              const float* __restrict__ hvec,
                                              const float* __restrict__ K,
                                              const float* __restrict__ R,
                                              const float* __restrict__ bias,
                                              float* __restrict__ z) {
    __shared__ float hs[C_DIM];
    __shared__ float xs[WC];
    __shared__ float red[256];
    int tid = threadIdx.x;
    for (int i = tid; i < C_DIM; i += 256) hs[i] = hvec[i];
    if (tid < WC) xs[tid] = xt[tid];
    __syncthreads();
    int tx = tid & 63, ty = tid >> 6;
    int j = blockIdx.x * 64 + tx;
    float p = 0.f;
    if (j < Z_DIM) {
        for (int i = ty; i < WC; i += 4)    p = fmaf(xs[i], K[(long)i * Z_DIM + j], p);
        for (int i = ty; i < C_DIM; i += 4) p = fmaf(hs[i], R[(long)i * Z_DIM + j], p);
    }
    red[tid] = p;
    __syncthreads();
    if (ty == 0 && j < Z_DIM)
        z[j] = red[tx] + red[tx + 64] + red[tx + 128] + red[tx + 192] + bias[j];
}

// ---------------- LSTM gates ------------------------------------------------
__global__ void k_gates(const float* __restrict__ z, float* h, float* c) {
    int i = blockIdx.x * 256 + threadIdx.x;
    if (i >= C_DIM) return;
    float zi = z[i], zf = z[C_DIM + i], zg = z[2 * C_DIM + i], zo = z[3 * C_DIM + i];
    float cn = sigf(zf) * c[i] + sigf(zi) * tanhf(zg);
    c[i] = cn;
    h[i] = sigf(zo) * tanhf(cn);
}

// ---------------- out = h@Wo ; iface[0:645] = h@Wi --------------------------
__global__ void k_outiface(const float* __restrict__ h, const float* __restrict__ Wo,
                           const float* __restrict__ Wi, float* __restrict__ out,
                           float* __restrict__ ifc) {
    __shared__ float hs[C_DIM];
    int tid = threadIdx.x;
    for (int i = tid; i < C_DIM; i += 256) hs[i] = h[i];
    __syncthreads();
    int j = blockIdx.x * 256 + tid;
    if (j < OUTD) {
        float acc = 0.f;
        for (int i = 0; i < C_DIM; ++i) acc = fmaf(hs[i], Wo[(long)i * OUTD + j], acc);
        out[j] = acc;
    } else if (j < OUTD + IFACE_N) {
        int jj = j - OUTD;
        float acc = 0.f;
        for (int i = 0; i < C_DIM; ++i) acc = fmaf(hs[i], Wi[(long)i * IFACE_D + jj], acc);
        ifc[jj] = acc;
    }
}

// ---------------- build padded key matrix (16x128): rows 0-3 read keys,
// row 4 write key, each l2-normalized and pre-scaled by beta; rows 5-15 zero --
__global__ void k_keys(const float* __restrict__ ifc, float* __restrict__ kmat) {
    int tid = threadIdx.x;
    int lane = tid & 31, wid = tid >> 5;
    if (wid < 5) {
        const float* src = (wid < 4) ? (ifc + wid * WC) : (ifc + 4 * WC + RHEADS);
        float v[4]; float ss = 0.f;
        #pragma unroll
        for (int q = 0; q < 4; ++q) { v[q] = src[lane * 4 + q]; ss = fmaf(v[q], v[q], ss); }
        for (int d = 16; d >= 1; d >>= 1) ss += __shfl_xor(ss, d, 32);
        float bin = (wid < 4) ? ifc[4 * WC + wid] : ifc[4 * WC + RHEADS + WC];
        float sp = (bin > 20.f) ? bin : log1pf(__expf(bin));      // softplus
        float scale = (1.f + sp) * rsqrtf(fmaxf(ss, 1e-12f));
        #pragma unroll
        for (int q = 0; q < 4; ++q) kmat[wid * WC + lane * 4 + q] = v[q] * scale;
    } else {
        for (int i = (wid - 5) * 32 + lane; i < 11 * WC; i += 96) kmat[5 * WC + i] = 0.f;
    }
}

// ---------------- content scores via V_WMMA_F32_16X16X4_F32 -----------------
// One wave = 16 rows of M x 16 key columns (5 valid). A fragment: lane l holds
// row (l&15), K pair {4j+2*(l>>4), +1} -> the same float2 feeds sum-of-squares.
// D is scaled per-row by rsqrt(||m||^2) afterwards (l2norm is a row scalar).
__global__ __launch_bounds__(256) void k_scores(const float* __restrict__ M,
                                                const float* __restrict__ kmat,
                                                float* __restrict__ wread,
                                                float* __restrict__ wwrite,
                                                float* __restrict__ colmax) {
    __shared__ float kk[16 * WC];
    __shared__ float wmax[8 * 5];
    int tid = threadIdx.x;
    for (int i = tid; i < 16 * WC; i += 256) kk[i] = kmat[i];
    __syncthreads();
    int lane = tid & 31, wid = tid >> 5;
    int half = lane >> 4;          // selects K pair within each 4-block
    int sub  = lane & 15;          // A: row index, B/D: column index
    long rowbase = (long)blockIdx.x * 128 + wid * 16;
    const float* arow = M + (rowbase + sub) * WC + 2 * half;
    const float* brow = kk + sub * WC + 2 * half;
    v8f acc = {};
    float ss = 0.f;
    #pragma unroll
    for (int j = 0; j < 32; ++j) {
        v2f a = *(const v2f*)(arow + 4 * j);
        v2f b = *(const v2f*)(brow + 4 * j);
        ss = fmaf(a.x, a.x, ss);
        ss = fmaf(a.y, a.y, ss);
        acc = __builtin_amdgcn_wmma_f32_16x16x4_f32(false, a, false, b,
                                                    (short)0, acc, false, false);
    }
    ss += __shfl_xor(ss, 16, 32);                 // combine the two half-rows
    float rnorm = rsqrtf(fmaxf(ss, 1e-12f));      // lane l: norm of row l&15
    float vals[8];
    float smax = -3.4e38f;
    #pragma unroll
    for (int v = 0; v < 8; ++v) {
        float rn = __shfl(rnorm, v + 8 * half, 32);   // norm of D row v+8*half
        vals[v] = acc[v] * rn;
        smax = fmaxf(smax, vals[v]);
    }
    int col = sub;
    if (col < RHEADS) {
        #pragma unroll
        for (int v = 0; v < 8; ++v) wread[(rowbase + v + 8 * half) * RHEADS + col] = vals[v];
    } else if (col == RHEADS) {
        #pragma unroll
        for (int v = 0; v < 8; ++v) wwrite[rowbase + v + 8 * half] = vals[v];
    }
    smax = fmaxf(smax, __shfl_xor(smax, 16, 32));
    if (lane < 5) wmax[wid * 5 + lane] = smax;
    __syncthreads();
    if (tid < 5) {
        float m = wmax[tid];
        for (int w = 1; w < 8; ++w) m = fmaxf(m, wmax[w * 5 + tid]);
        atomicMaxF(colmax + tid, m);
    }
}

// ---------------- softmax pass 2: exp(s - max) + column sums ----------------
__global__ void k_expsum(float* __restrict__ wread, float* __restrict__ wwrite,
                         const float* __restrict__ colmax, float* __restrict__ colsum) {
    __shared__ float bs[5];
    int tid = threadIdx.x;
    if (tid < 5) bs[tid] = 0.f;
    __syncthreads();
    float cm0 = colmax[0], cm1 = colmax[1], cm2 = colmax[2], cm3 = colmax[3], cm4 = colmax[4];
    float l0 = 0.f, l1 = 0.f, l2 = 0.f, l3 = 0.f, l4 = 0.f;
    int base = blockIdx.x * 256 + tid;
    for (int kk = 0; kk < 4; ++kk) {
        int n = base + kk * 65536;
        float e;
        e = __expf(wread[n * 4 + 0] - cm0); wread[n * 4 + 0] = e; l0 += e;
        e = __expf(wread[n * 4 + 1] - cm1); wread[n * 4 + 1] = e; l1 += e;
        e = __expf(wread[n * 4 + 2] - cm2); wread[n * 4 + 2] = e; l2 += e;
        e = __expf(wread[n * 4 + 3] - cm3); wread[n * 4 + 3] = e; l3 += e;
        e = __expf(wwrite[n]        - cm4); wwrite[n]        = e; l4 += e;
    }
    for (int d = 16; d >= 1; d >>= 1) {
        l0 += __shfl_xor(l0, d, 32); l1 += __shfl_xor(l1, d, 32);
        l2 += __shfl_xor(l2, d, 32); l3 += __shfl_xor(l3, d, 32);
        l4 += __shfl_xor(l4, d, 32);
    }
    if ((tid & 31) == 0) {
        atomicAdd(&bs[0], l0); atomicAdd(&bs[1], l1); atomicAdd(&bs[2], l2);
        atomicAdd(&bs[3], l3); atomicAdd(&bs[4], l4);
    }
    __syncthreads();
    if (tid < 5) atomicAdd(colsum + tid, bs[tid]);
}

// ---------------- softmax pass 3: normalize ---------------------------------
__global__ void k_norm(float* __restrict__ wread, float* __restrict__ wwrite,
                       const float* __restrict__ colsum) {
    int n = blockIdx.x * 256 + threadIdx.x;
    float i0 = 1.f / colsum[0], i1 = 1.f / colsum[1], i2 = 1.f / colsum[2];
    float i3 = 1.f / colsum[3], i4 = 1.f / colsum[4];
    wread[n * 4 + 0] *= i0; wread[n * 4 + 1] *= i1;
    wread[n * 4 + 2] *= i2; wread[n * 4 + 3] *= i3;
    wwrite[n] *= i4;
}

// ---------------- allocation: sort + cumprod + scatter ----------------------
__global__ void k_makekeys(const float* __restrict__ usage, u64* __restrict__ keys) {
    int n = blockIdx.x * 256 + threadIdx.x;
    keys[n] = ((u64)__float_as_uint(usage[n]) << 32) | (unsigned)n;
}

// bitonic: fully sort 2048-chunks in LDS (stages k=2..2048)
__global__ __launch_bounds__(1024) void k_bsort_local(u64* __restrict__ keys) {
    __shared__ u64 sh[2048];
    int t = threadIdx.x;
    int base = blockIdx.x * 2048;
    sh[t] = keys[base + t];
    sh[t + 1024] = keys[base + t + 1024];
    __syncthreads();
    for (int k = 2; k <= 2048; k <<= 1) {
        for (int j = k >> 1; j > 0; j >>= 1) {
            int i = ((t & ~(j - 1)) << 1) | (t & (j - 1));
            int p = i | j;
            bool asc = (((unsigned)(base + i)) & (unsigned)k) == 0;
            u64 a = sh[i], b = sh[p];
            if ((a > b) == asc) { sh[i] = b; sh[p] = a; }
            __syncthreads();
        }
    }
    keys[base + t] = sh[t];
    keys[base + t + 1024] = sh[t + 1024];
}

// bitonic global step (j >= 2048)
__global__ void k_bsort_global(u64* __restrict__ keys, int j, int k) {
    int t = blockIdx.x * 256 + threadIdx.x;       // N/2 threads
    int i = ((t & ~(j - 1)) << 1) | (t & (j - 1));
    int p = i | j;
    bool asc = ((unsigned)i & (unsigned)k) == 0;
    u64 a = keys[i], b = keys[p];
    if ((a > b) == asc) { keys[i] = b; keys[p] = a; }
}

// bitonic tail merge (j=1024..1) in LDS for stage k
__global__ __launch_bounds__(1024) void k_bsort_localmerge(u64* __restrict__ keys, int k) {
    __shared__ u64 sh[2048];
    int t = threadIdx.x;
    int base = blockIdx.x * 2048;
    sh[t] = keys[base + t];
    sh[t + 1024] = keys[base + t + 1024];
    __syncthreads();
    bool asc = (((unsigned)base) & (unsigned)k) == 0;   // uniform: k >= 4096 > 2048
    for (int j = 1024; j > 0; j >>= 1) {
        int i = ((t & ~(j - 1)) << 1) | (t & (j - 1));
        int p = i | j;
        u64 a = sh[i], b = sh[p];
        if ((a > b) == asc) { sh[i] = b; sh[p] = a; }
        __syncthreads();
    }
    keys[base + t] = sh[t];
    keys[base + t + 1024] = sh[t + 1024];
}

__global__ void k_extract(const u64* __restrict__ keys, float* __restrict__ s,
                          unsigned* __restrict__ idxs) {
    int n = blockIdx.x * 256 + threadIdx.x;
    u64 kv = keys[n];
    s[n] = __uint_as_float((unsigned)(kv >> 32));
    idxs[n] = (unsigned)kv;
}

// inclusive product scan within 2048-chunks (shfl-based, no division)
__global__ __launch_bounds__(1024) void k_scan1(const float* __restrict__ s,
                                                float* __restrict__ P,
                                                float* __restrict__ btot) {
    __shared__ float woff[32];
    int t = threadIdx.x;
    int base = blockIdx.x * 2048;
    float a0 = s[base + 2 * t], a1 = s[base + 2 * t + 1];
    float tp = a0 * a1;
    int lane = t & 31, wid = t >> 5;
    float v = tp;
    for (int d = 1; d < 32; d <<= 1) {
        float up = __shfl_up(v, d, 32);
        if (lane >= d) v *= up;
    }
    if (lane == 31) woff[wid] = v;
    __syncthreads();
    if (t < 32) {
        float wv = woff[t];
        for (int d = 1; d < 32; d <<= 1) {
            float up = __shfl_up(wv, d, 32);
            if (t >= d) wv *= up;
        }
        woff[t] = wv;
    }
    __syncthreads();
    float waveExcl = (wid == 0) ? 1.f : woff[wid - 1];
    float prevIncl = __shfl_up(v, 1, 32);
    float threadExcl = waveExcl * ((lane == 0) ? 1.f : prevIncl);
    P[base + 2 * t]     = threadExcl * a0;
    P[base + 2 * t + 1] = threadExcl * tp;
    if (t == 1023) btot[blockIdx.x] = waveExcl * v;
}

__global__ void k_scan2(const float* __restrict__ btot, float* __restrict__ bpref) {
    float p = 1.f;
    for (int i = 0; i < 128; ++i) { bpref[i] = p; p *= btot[i]; }
}

__global__ void k_alloc(const float* __restrict__ s, const unsigned* __restrict__ idxs,
                        const float* __restrict__ P, const float* __restrict__ bpref,
                        float* __restrict__ alloc) {
    int i = blockIdx.x * 256 + threadIdx.x;
    int b = i >> 11, pos = i & 2047;
    float cpe = bpref[b] * (pos ? P[i - 1] : 1.f);    // exclusive cumprod
    alloc[idxs[i]] = (1.f - s[i]) * cpe;
}

// ---------------- launcher --------------------------------------------------
extern "C" void kernel_launch(void* const* d_in, const int* in_sizes, int n_in,
                              void* d_out, int out_size, void* d_ws, size_t ws_size,
                              hipStream_t stream) {
    const float* x     = (const float*)d_in[0];
    const float* Wd    = (const float*)d_in[1];
    const float* bd    = (const float*)d_in[2];
    const float* Klstm = (const float*)d_in[3];
    const float* Rlstm = (const float*)d_in[4];
    const float* bL    = (const float*)d_in[5];
    const float* h0    = (const float*)d_in[6];
    const float* c0    = (const float*)d_in[7];
    const float* readv = (const float*)d_in[8];
    const float* Wo    = (const float*)d_in[9];
    const float* Wi    = (const float*)d_in[10];
    const float* Mmem  = (const float*)d_in[11];
    const float* usage = (const float*)d_in[12];

    float* out    = (float*)d_out;
    float* wread  = out + OUTD;
    float* wwrite = out + OUTD + 4 * (size_t)N_TOT;
    float* alloc  = out + OUTD + 5 * (size_t)N_TOT;

    char* ws = (char*)d_ws;
    u64*      keys     = (u64*)ws;                                  // N * 8B
    float*    s_sorted = (float*)(ws + (size_t)N_TOT * 8);          // N * 4B
    unsigned* idxs     = (unsigned*)(ws + (size_t)N_TOT * 12);      // N * 4B
    float*    P        = (float*)(ws + (size_t)N_TOT * 16);         // N * 4B
    float*    misc     = (float*)(ws + (size_t)N_TOT * 20);
    float* h      = misc;               // 1431 (pad 1536)
    float* c      = misc + 1536;        // 1431 (pad -> 3072)
    float* xw     = misc + 3072;        // 128  (pad -> 3328)
    float* z      = misc + 3328;        // 5724 (pad -> 9216)
    float* ifc    = misc + 9216;        // 645  (pad -> 9984)
    float* kmat   = misc + 9984;        // 2048 -> 12032
    float* colmax = misc + 12032;       // 5 (pad 8)
    float* colsum = colmax + 8;         // 5 (pad 8)
    float* btot   = colmax + 16;        // 128
    float* bpref  = btot + 128;         // 128

    // ---- front end: dense + 5 LSTM steps (sequential by stream order) ----
    k_init<<<6, 256, 0, stream>>>(x, Wd, bd, h0, c0, h, c, xw, colmax, colsum);
    for (int t = 0; t < 5; ++t) {
        const float* xt = (t == 0) ? xw : (readv + (t - 1) * WC);
        k_zmat<<<90, 256, 0, stream>>>(xt, h, Klstm, Rlstm, bL, z);
        k_gates<<<6, 256, 0, stream>>>(z, h, c);
    }
    k_outiface<<<5, 256, 0, stream>>>(h, Wo, Wi, out, ifc);
    k_keys<<<1, 256, 0, stream>>>(ifc, kmat);

    // ---- content addressing: WMMA scores + 3-pass column softmax over N ----
    k_scores<<<N_TOT / 128, 256, 0, stream>>>(Mmem, kmat, wread, wwrite, colmax);
    k_expsum<<<256, 256, 0, stream>>>(wread, wwrite, colmax, colsum);
    k_norm<<<N_TOT / 256, 256, 0, stream>>>(wread, wwrite, colsum);

    // ---- allocation weighting: u64 bitonic sort + product scan + scatter ----
    k_makekeys<<<N_TOT / 256, 256, 0, stream>>>(usage, keys);
    k_bsort_local<<<N_TOT / 2048, 1024, 0, stream>>>(keys);
    for (int k = 4096; k <= N_TOT; k <<= 1) {
        for (int j = k >> 1; j >= 2048; j >>= 1)
            k_bsort_global<<<N_TOT / 512, 256, 0, stream>>>(keys, j, k);
        k_bsort_localmerge<<<N_TOT / 2048, 1024, 0, stream>>>(keys, k);
    }
    k_extract<<<N_TOT / 256, 256, 0, stream>>>(keys, s_sorted, idxs);
    k_scan1<<<N_TOT / 2048, 1024, 0, stream>>>(s_sorted, P, btot);
    k_scan2<<<1, 1, 0, stream>>>(btot, bpref);
    k_alloc<<<N_TOT / 256, 256, 0, stream>>>(s_sorted, idxs, P, bpref, alloc);
}